// EdgeAggregationLayer_59184649339042
// MI455X (gfx1250) — compile-verified
//
#include <hip/hip_runtime.h>

typedef float v2f __attribute__((ext_vector_type(2)));
typedef float v4f __attribute__((ext_vector_type(4)));
typedef float v8f __attribute__((ext_vector_type(8)));

#define IN_C   128
#define EDGE_C 64
#define OUT_C  64

// ---------------------------------------------------------------------------
// Stage 0: fuse the two bias-free linears: Wc[o][i] = sum_j W2[o][j]*W1[j][i]
// 64x128 output, one block, negligible cost.
// ---------------------------------------------------------------------------
__global__ void ea_fuse_weights(const float* __restrict__ W1,
                                const float* __restrict__ W2,
                                float* __restrict__ Wc) {
  for (int t = threadIdx.x; t < OUT_C * IN_C; t += 256) {
    const int o = t >> 7;    // / 128
    const int i = t & 127;   // % 128
    float s = 0.0f;
#pragma unroll
    for (int j = 0; j < EDGE_C; ++j)
      s = fmaf(W2[o * EDGE_C + j], W1[j * IN_C + i], s);
    Wc[t] = s;
  }
}

// ---------------------------------------------------------------------------
// Stage 1: y[n][o] = dot(x[n][:], Wc[o][:])  -- per-NODE GEMM, fp32 WMMA.
// Wave-per-16-row tile, 4 column tiles (OUT_C = 64), K = 128 in steps of 4
// via V_WMMA_F32_16X16X4_F32 (full fp32, no precision loss vs reference).
//
// A (16x4 f32) lane layout (ISA 7.12.2): lanes 0-15 row M=lane hold K=k,k+1;
// lanes 16-31 hold K=k+2,k+3  -> contiguous float2 from the x row.
// B (4x16) mirrors it (N across lanes) -> contiguous float2 from a Wc row.
// ---------------------------------------------------------------------------
__global__ void ea_node_gemm_wmma(const float* __restrict__ x,
                                  const float* __restrict__ Wc,
                                  float* __restrict__ y,
                                  int n_nodes) {
  const int lane    = threadIdx.x & 31;
  const int wave    = threadIdx.x >> 5;
  const int rowBase = (blockIdx.x * 8 + wave) * 16;
  if (rowBase >= n_nodes) return;      // whole-wave uniform exit, EXEC stays full

  const int half = lane >> 4;          // 0: K=k,k+1   1: K=k+2,k+3
  const int l16  = lane & 15;

  // clamp so partial last tile reads stay in bounds (stores are masked below)
  const int rA = min(rowBase + l16, n_nodes - 1);
  const float* __restrict__ xrow = x + (size_t)rA * IN_C + 2 * half;
  const float* __restrict__ w0 = Wc + (size_t)(0 * 16 + l16) * IN_C + 2 * half;
  const float* __restrict__ w1 = Wc + (size_t)(1 * 16 + l16) * IN_C + 2 * half;
  const float* __restrict__ w2 = Wc + (size_t)(2 * 16 + l16) * IN_C + 2 * half;
  const float* __restrict__ w3 = Wc + (size_t)(3 * 16 + l16) * IN_C + 2 * half;

  v8f acc0 = {}, acc1 = {}, acc2 = {}, acc3 = {};

#pragma unroll 4
  for (int k = 0; k < IN_C; k += 4) {
    const v2f a  = *(const v2f*)(xrow + k);
    const v2f b0 = *(const v2f*)(w0 + k);
    const v2f b1 = *(const v2f*)(w1 + k);
    const v2f b2 = *(const v2f*)(w2 + k);
    const v2f b3 = *(const v2f*)(w3 + k);
    acc0 = __builtin_amdgcn_wmma_f32_16x16x4_f32(false, a, false, b0, (short)0, acc0, false, false);
    acc1 = __builtin_amdgcn_wmma_f32_16x16x4_f32(false, a, false, b1, (short)0, acc1, false, false);
    acc2 = __builtin_amdgcn_wmma_f32_16x16x4_f32(false, a, false, b2, (short)0, acc2, false, false);
    acc3 = __builtin_amdgcn_wmma_f32_16x16x4_f32(false, a, false, b3, (short)0, acc3, false, false);
  }

  // D layout: VGPR v holds (M = v + 8*half, N = l16)
#pragma unroll
  for (int v = 0; v < 8; ++v) {
    const int row = rowBase + v + 8 * half;
    if (row < n_nodes) {
      float* yr = y + (size_t)row * OUT_C + l16;
      yr[0]  = acc0[v];
      yr[16] = acc1[v];
      yr[32] = acc2[v];
      yr[48] = acc3[v];
    }
  }
}

// ---------------------------------------------------------------------------
// Stage 2: out[e][:] = y[row[e]][:].  16 lanes per edge, float4 (b128) each.
// y reads are regular-temporal (footprint 25.6 MB -> L2-resident);
// out stores are non-temporal (409.6 MB streaming write, keep L2 for y).
// ---------------------------------------------------------------------------
__global__ void ea_gather_out(const float* __restrict__ y,
                              const int* __restrict__ rowidx,
                              float* __restrict__ out,
                              int n_edges, int n_nodes) {
  __shared__ int rows_s[16];
  const int e0 = blockIdx.x * 16;

  if (threadIdx.x < 16) {
    const int e = e0 + threadIdx.x;
    int r = (e < n_edges) ? rowidx[e] : 0;
    if ((unsigned)r >= (unsigned)n_nodes) r = 0;   // defensive clamp
    rows_s[threadIdx.x] = r;
  }
  __syncthreads();

  const int g = threadIdx.x >> 4;   // edge slot within block
  const int q = threadIdx.x & 15;   // float4 chunk within edge row
  const int e = e0 + g;
  if (e < n_edges) {
    const int r = rows_s[g];
    const v4f v = *(const v4f*)(y + (size_t)r * OUT_C + q * 4);
    __builtin_nontemporal_store(v, (v4f*)(out + (size_t)e * OUT_C + q * 4));
  }
}

// ---------------------------------------------------------------------------
extern "C" void kernel_launch(void* const* d_in, const int* in_sizes, int n_in,
                              void* d_out, int out_size, void* d_ws, size_t ws_size,
                              hipStream_t stream) {
  const float* x  = (const float*)d_in[0];
  const int*   ei = (const int*)d_in[1];   // edge_index [2, E]; row = ei[0..E)
  const float* W1 = (const float*)d_in[2]; // [EDGE_C, IN_C]
  const float* W2 = (const float*)d_in[3]; // [OUT_C, EDGE_C]
  float* out = (float*)d_out;

  const int n_nodes = in_sizes[0] / IN_C;
  const int n_edges = in_sizes[1] / 2;

  float* Wc = (float*)d_ws;                                     // 32 KB
  float* y  = (float*)((char*)d_ws + OUT_C * IN_C * sizeof(float)); // N*64*4 = 25.6 MB

  ea_fuse_weights<<<1, 256, 0, stream>>>(W1, W2, Wc);

  const int gemm_blocks = (n_nodes + 127) / 128;                // 8 waves x 16 rows
  ea_node_gemm_wmma<<<gemm_blocks, 256, 0, stream>>>(x, Wc, y, n_nodes);

  const int gather_blocks = (n_edges + 15) / 16;                // 16 edges / block
  ea_gather_out<<<gather_blocks, 256, 0, stream>>>(y, ei, out, n_edges, n_nodes);
}